// LearnablePermutation_49993419325966
// MI455X (gfx1250) — compile-verified
//
#include <hip/hip_runtime.h>

// Gumbel-Sinkhorn soft permutation, LDS-resident per 128x128 batch matrix.
// gamma: [128,128] f32 (L2-resident, 64KB), noise: [4096,128,128] f32,
// out = exp(sinkhorn((gamma+noise)/0.2, 20 iters)).
//
// CDNA5 path: GLOBAL_LOAD_ASYNC_TO_LDS_B128 scatters the tile straight into a
// padded (stride-129) LDS layout (bank-conflict-free for both row and column
// passes), synced with s_wait_asynccnt. All 20 Sinkhorn iterations run out of
// LDS; HBM sees exactly one read + one write of the 512MB working set
// (~22us floor at 23.3 TB/s). WMMA is deliberately not used: the workload has
// no GEMM structure (only 128-wide max/exp-sum reductions), so v_exp_f32 VALU
// + DS is the roofline-correct instruction mix.

#define N      128
#define NPAD   129            // +1 float row padding: bank = (r + c) & 63
#define INV_T  5.0f           // 1 / SINKHORN_TEMP

__global__ __launch_bounds__(128)
void sinkhorn_gumbel_kernel(const float* __restrict__ gamma,
                            const float* __restrict__ noise,
                            const int*   __restrict__ n_iters_p,
                            float*       __restrict__ out)
{
    __shared__ float tile[N * NPAD];       // 66048 B, sole LDS object -> base 0

    const int tid = threadIdx.x;           // 0..127
    const int b   = blockIdx.x;            // batch matrix index
    const int n_iters = *n_iters_p;

    // ---------------- async copy: global noise tile -> padded LDS ----------
    // Each lane issues 32 x B128 async loads; LDS dest address carries the
    // row padding so no shuffle pass is needed afterwards.
    const unsigned long long src_base =
        (unsigned long long)(const void*)noise +
        (unsigned long long)b * (unsigned long long)(N * N * sizeof(float));

    #pragma unroll
    for (int i = 0; i < (N * N) / (4 * N); ++i) {   // 32 float4 per thread
        const int v = i * N + tid;                  // float4 index in matrix
        const int e = v * 4;                        // element index
        const int r = e >> 7;
        const int c = e & (N - 1);
        const unsigned lds_off = (unsigned)(r * NPAD + c) * 4u;
        const unsigned g_off   = (unsigned)e * 4u;
        asm volatile("global_load_async_to_lds_b128 %0, %1, %2"
                     :
                     : "v"(lds_off), "v"(g_off), "s"(src_base)
                     : "memory");
    }
    asm volatile("s_wait_asynccnt 0" ::: "memory");
    __syncthreads();

    // ---------------- fuse (x + gamma) * inv_temp (coalesced gamma reads) --
    const float4* __restrict__ g4 = (const float4*)gamma;
    #pragma unroll
    for (int i = 0; i < (N * N) / (4 * N); ++i) {
        const int v = i * N + tid;
        const int e = v * 4;
        const int r = e >> 7;
        const int c = e & (N - 1);
        const float4 g = g4[v];
        float* p = &tile[r * NPAD + c];
        p[0] = (p[0] + g.x) * INV_T;
        p[1] = (p[1] + g.y) * INV_T;
        p[2] = (p[2] + g.z) * INV_T;
        p[3] = (p[3] + g.w) * INV_T;
    }
    __syncthreads();

    // ---------------- Sinkhorn iterations, fully LDS-resident --------------
    for (int it = 0; it < n_iters; ++it) {
        {   // row logsumexp: thread tid owns row tid
            float* __restrict__ row = &tile[tid * NPAD];
            float m = -3.0e38f;
            #pragma unroll 8
            for (int c = 0; c < N; ++c) m = fmaxf(m, row[c]);
            float s = 0.0f;
            #pragma unroll 8
            for (int c = 0; c < N; ++c) s += __expf(row[c] - m);
            const float lse = m + __logf(s);
            #pragma unroll 8
            for (int c = 0; c < N; ++c) row[c] -= lse;
        }
        __syncthreads();
        {   // col logsumexp: thread tid owns column tid
            float* __restrict__ col = &tile[tid];
            float m = -3.0e38f;
            #pragma unroll 8
            for (int r = 0; r < N; ++r) m = fmaxf(m, col[r * NPAD]);
            float s = 0.0f;
            #pragma unroll 8
            for (int r = 0; r < N; ++r) s += __expf(col[r * NPAD] - m);
            const float lse = m + __logf(s);
            #pragma unroll 8
            for (int r = 0; r < N; ++r) col[r * NPAD] -= lse;
        }
        __syncthreads();
    }

    // ---------------- fused exp + coalesced float4 store -------------------
    float4* __restrict__ o4 = (float4*)out + (size_t)b * (size_t)(N * N / 4);
    #pragma unroll
    for (int i = 0; i < (N * N) / (4 * N); ++i) {
        const int v = i * N + tid;
        const int e = v * 4;
        const int r = e >> 7;
        const int c = e & (N - 1);
        const float* p = &tile[r * NPAD + c];
        float4 val;
        val.x = __expf(p[0]);
        val.y = __expf(p[1]);
        val.z = __expf(p[2]);
        val.w = __expf(p[3]);
        o4[v] = val;
    }
}

extern "C" void kernel_launch(void* const* d_in, const int* in_sizes, int n_in,
                              void* d_out, int out_size, void* d_ws, size_t ws_size,
                              hipStream_t stream)
{
    const float* gamma = (const float*)d_in[0];            // [128,128] f32
    const float* noise = (const float*)d_in[1];            // [B,128,128] f32
    const int*   iters = (const int*)d_in[2];              // scalar
    float*       out   = (float*)d_out;                    // [B,128,128] f32

    const int B = in_sizes[1] / (N * N);                   // 4096
    sinkhorn_gumbel_kernel<<<B, N, 0, stream>>>(gamma, noise, iters, out);
}